// PerseverativeAgent_11639361372462
// MI455X (gfx1250) — compile-verified
//
#include <hip/hip_runtime.h>
#include <hip/hip_bf16.h>

typedef __attribute__((ext_vector_type(2))) float v2f;
typedef __attribute__((ext_vector_type(8))) float v8f;

#define N_SESS   2048
#define N_TRIALS 4096
#define N_VIN    16      // x innermost dim = 2*n_vals
#define N_VALS   8
#define CHUNK    16
#define N_CHUNKS (N_TRIALS / CHUNK)   // 256
#define WAVES_PER_BLOCK 8
#define N_TILES  (N_SESS / 2)         // 16 series per tile = 2 sessions x 8 vals

// EMA as blocked scan: per chunk of 16 trials,
//   out = L @ X  +  d (x) carry
// L[t,k] = alpha*beta^(t-k) (k<=t), d[t] = beta^(t+1).
// L @ X via 4 accumulating V_WMMA_F32_16X16X4_F32 (K=16 = 4x4), fp32 end to end.
// Carry rank-1 term applied AFTER the WMMA chain (FMA) so the loop-carried
// serial chain is just bpermute -> fma(c[7]) -> bpermute; B loads are
// register double-buffered so WMMAs never wait on same-iteration loads.
__global__ __launch_bounds__(WAVES_PER_BLOCK * 32)
void ema_wmma_kernel(const float* __restrict__ x,
                     const float* __restrict__ alpha_raw,
                     float* __restrict__ out)
{
    const int lane = threadIdx.x & 31;
    const int tile = blockIdx.x * WAVES_PER_BLOCK + (threadIdx.x >> 5);

    // alpha = sigmoid(alpha_raw[0]); uniform across the grid
    const float ar    = alpha_raw[0];
    const float alpha = 1.0f / (1.0f + __expf(-ar));
    const float beta  = 1.0f - alpha;

    const int m    = lane & 15;   // A-matrix row
    const int half = lane >> 4;   // K-half selector for A/B, M-half for C/D

    // ---- Build the 4 A-matrices (lower-triangular EMA kernel), once per wave ----
    // A_j lane layout (32-bit A 16x4): lanes 0-15: K = 4j+{0,1}; lanes 16-31: K = 4j+{2,3}
    v2f A[4];
#pragma unroll
    for (int j = 0; j < 4; ++j) {
#pragma unroll
        for (int v = 0; v < 2; ++v) {
            const int k = 4 * j + 2 * half + v;
            float val = 0.0f;
            if (k <= m) {
                val = alpha;
                for (int i = 0; i < m - k; ++i) val *= beta;  // alpha * beta^(m-k)
            }
            A[j][v] = val;
        }
    }

    // ---- Per-lane decay vector d[t] = beta^(t+1), t = r + 8*half ----
    float dr[8];
    {
        const float b2 = beta * beta, b4 = b2 * b2, b8 = b4 * b4;
        float p = half ? (b8 * beta) : beta;   // beta^(8*half + 1)
#pragma unroll
        for (int r = 0; r < 8; ++r) { dr[r] = p; p *= beta; }
    }

    // ---- Column -> (session, value) mapping; both lane-halves share columns ----
    const int n    = lane & 15;
    const int sess = 2 * tile + (n >> 3);
    const int val  = n & 7;
    const float* xcol = x   + (size_t)sess * N_TRIALS * N_VIN  + val;  // x[sess][t][val],  t-stride 16
    float*       ocol = out + (size_t)sess * N_TRIALS * N_VALS + val;  // out[sess][t][val], t-stride 8

    float carry = 0.0f;   // state_{-1} = 0

    // ---- Prologue: load B for chunk 0 ----
    v2f Bc[4];
#pragma unroll
    for (int j = 0; j < 4; ++j) {
        const int t0 = 4 * j + 2 * half;
        Bc[j][0] = xcol[(size_t)t0       * N_VIN];
        Bc[j][1] = xcol[(size_t)(t0 + 1) * N_VIN];
    }

    for (int chunk = 0; chunk < N_CHUNKS; ++chunk) {
        const int tb = chunk * CHUNK;

        // Warm L2/WGP$ two chunks ahead (global_prefetch_b8, WGP scope)
        if (chunk + 2 < N_CHUNKS)
            __builtin_prefetch(&xcol[(size_t)(tb + 2 * CHUNK) * N_VIN], 0, 3);

        // ---- Issue next chunk's B loads (register double-buffer) ----
        const int tbn = (chunk + 1 < N_CHUNKS ? tb + CHUNK : tb);  // clamp: last iter reloads (discarded)
        v2f Bn[4];
#pragma unroll
        for (int j = 0; j < 4; ++j) {
            const int t0 = tbn + 4 * j + 2 * half;
            Bn[j][0] = xcol[(size_t)t0       * N_VIN];
            Bn[j][1] = xcol[(size_t)(t0 + 1) * N_VIN];
        }

        // ---- Y = L @ X : 4 accumulating f32 WMMAs from C = 0 (carry-independent) ----
        v8f c;
#pragma unroll
        for (int r = 0; r < 8; ++r) c[r] = 0.0f;
#pragma unroll
        for (int j = 0; j < 4; ++j)
            c = __builtin_amdgcn_wmma_f32_16x16x4_f32(
                    /*neg_a=*/false, A[j], /*neg_b=*/false, Bc[j],
                    /*c_mod=*/(short)0, c, /*reuse_a=*/false, /*reuse_b=*/false);

        // ---- Add rank-1 carry term: c[r] += beta^(t+1) * carry[n] ----
#pragma unroll
        for (int r = 0; r < 8; ++r) c[r] = __builtin_fmaf(dr[r], carry, c[r]);

        // ---- Store chunk: C layout VGPR r, lane l -> (t = r + 8*half, n) ----
#pragma unroll
        for (int r = 0; r < 8; ++r) {
            const int t = tb + r + 8 * half;
            ocol[(size_t)t * N_VALS] = c[r];
        }

        // ---- Carry handoff: row t=15 lives in c[7] of lanes 16..31 ----
        carry = __shfl(c[7], 16 + n, 32);

        // rotate double buffer
#pragma unroll
        for (int j = 0; j < 4; ++j) Bc[j] = Bn[j];
    }
}

extern "C" void kernel_launch(void* const* d_in, const int* in_sizes, int n_in,
                              void* d_out, int out_size, void* d_ws, size_t ws_size,
                              hipStream_t stream) {
    const float* x         = (const float*)d_in[0];
    const float* alpha_raw = (const float*)d_in[1];
    // d_in[2] = n_vals (compile-time constant 8 here)
    float* out = (float*)d_out;

    const int blocks  = N_TILES / WAVES_PER_BLOCK;  // 128
    const int threads = WAVES_PER_BLOCK * 32;       // 256
    ema_wmma_kernel<<<blocks, threads, 0, stream>>>(x, alpha_raw, out);
}